// GATClassifier_31482110280387
// MI455X (gfx1250) — compile-verified
//
#include <hip/hip_runtime.h>
#include <hip/hip_bf16.h>

typedef __attribute__((ext_vector_type(2))) float v2f;
typedef __attribute__((ext_vector_type(8))) float v8f;

#define NEG_SLOPE 0.2f
#define NNODES 4096
#define NEDGES 32768
#define ETOT (NEDGES + NNODES)
#define PSTRIDE 160   // floats per K-pair row: 128 data + 32 pad (bank-conflict-free b64)

// ---------------------------------------------------------------------------
// fp32 WMMA GEMM: C[M,N] = A[M,K] @ B[K,N] (+bias, +relu optional)
// Block = 256 threads (8 waves) -> 256x64 output tile.
// Wave  -> 32x64: 2 M-subtiles x 4 N-fragments = 8 v_wmma per K-step, fed by
//          2 global b64 A loads + 4 LDS b64 B reads (no repacking).
// B is staged through LDS in 32x64 chunks, K-pair interleaved:
//   (k,c) -> Bs[(k>>1)*PSTRIDE + c*2 + (k&1)]
// so a B fragment (B[kk][c], B[kk+1][c]) is a single aligned ds_load_b64.
// PSTRIDE=160 puts the two half-wave K-phases on disjoint LDS banks.
// Requires: M%256==0, N%64==0, K%32==0 (all call sites satisfy this).
// WMMA f32 16x16x4 layout (ISA 7.12.2):
//   A (16x4): lane m=L&15; lanes<16 hold K+0/K+1, lanes>=16 hold K+2/K+3
//   B (4x16): lane n=L&15; same K split as A
//   C/D:      lane n=L&15; vgpr r holds row r + 8*(L>=16)
// ---------------------------------------------------------------------------
__global__ __launch_bounds__(256) void gemm_wmma_f32(
    const float* __restrict__ A, const float* __restrict__ B,
    const float* __restrict__ bias, float* __restrict__ C,
    int M, int N, int K, int do_relu)
{
  __shared__ float Bs[16 * PSTRIDE];   // 32 K-rows as 16 interleaved pairs

  const int tid  = threadIdx.x;
  const int lane = tid & 31;
  const int wave = tid >> 5;
  const int tilesN = N >> 6;
  const int bm = blockIdx.x / tilesN;        // 256-row block index
  const int tn = blockIdx.x % tilesN;        // 64-col block index

  const int n  = lane & 15;
  const int kq = (lane >> 4) << 1;           // 0 or 2 (K phase per half-wave)

  const int row0 = bm * 256 + wave * 32 + (lane & 15);
  const float* Arow0 = A + (size_t)row0 * K;
  const float* Arow1 = Arow0 + (size_t)16 * K;

  // B-stage addressing: thread t holds rows (kk,kk+1) x cols (cc,cc+1)
  const int kk = (tid >> 5) * 2;             // 0,2,...,14
  const int cc = (tid & 31) * 2;             // 0,2,...,62
  const float* Bsrc = B + (size_t)kk * N + tn * 64 + cc;
  float* Bdst = &Bs[(kk >> 1) * PSTRIDE + cc * 2];

  v8f acc[2][4];
#pragma unroll
  for (int mi = 0; mi < 2; ++mi)
#pragma unroll
    for (int j = 0; j < 4; ++j) acc[mi][j] = (v8f){};

  for (int kc = 0; kc < K; kc += 32) {
    __syncthreads();                                   // LDS reads done
    {
      const float2 r0 = *(const float2*)(Bsrc + (size_t)kc * N);
      const float2 r1 = *(const float2*)(Bsrc + (size_t)(kc + 1) * N);
      float4 wv; wv.x = r0.x; wv.y = r1.x; wv.z = r0.y; wv.w = r1.y;
      *(float4*)Bdst = wv;                             // interleaved b128 store
    }
    __syncthreads();                                   // LDS chunk ready

#pragma unroll
    for (int k = 0; k < 32; k += 4) {
      const float2 t0 = *(const float2*)(Arow0 + kc + k + kq);
      const float2 t1 = *(const float2*)(Arow1 + kc + k + kq);
      v2f a0, a1;
      a0.x = t0.x; a0.y = t0.y;
      a1.x = t1.x; a1.y = t1.y;
      const int prow = (k >> 1) + (kq >> 1);           // K-pair row index
#pragma unroll
      for (int j = 0; j < 4; ++j) {
        const v2f b = *(const v2f*)&Bs[prow * PSTRIDE + (j * 16 + n) * 2];
        acc[0][j] = __builtin_amdgcn_wmma_f32_16x16x4_f32(
            false, a0, false, b, (short)0, acc[0][j], false, false);
        acc[1][j] = __builtin_amdgcn_wmma_f32_16x16x4_f32(
            false, a1, false, b, (short)0, acc[1][j], false, false);
      }
    }
  }

  const int colb = tn * 64 + n;
#pragma unroll
  for (int mi = 0; mi < 2; ++mi) {
    const int rbase = bm * 256 + wave * 32 + mi * 16 + ((lane >> 4) << 3);
#pragma unroll
    for (int r = 0; r < 8; ++r) {
      float* Crow = C + (size_t)(rbase + r) * N + colb;
#pragma unroll
      for (int j = 0; j < 4; ++j) {
        float v = acc[mi][j][r];
        if (bias) v += bias[tn * 64 + j * 16 + n];
        if (do_relu) v = v > 0.f ? v : 0.f;
        Crow[j * 16] = v;
      }
    }
  }
}

// ---------------------------------------------------------------------------
// alpha_src[n,h] = sum_c h[n,h,c]*a_src[h,c]; likewise alpha_dst. One wave per
// (n,h) row, 32-lane strided reduce + xor-shuffle tree (wave32).
// ---------------------------------------------------------------------------
__global__ __launch_bounds__(256) void alpha_kernel(
    const float* __restrict__ h, const float* __restrict__ a_src,
    const float* __restrict__ a_dst, float* __restrict__ out_src,
    float* __restrict__ out_dst, int NH, int H, int C)
{
  const int row = blockIdx.x * 8 + (threadIdx.x >> 5);
  if (row >= NH) return;
  const int lane = threadIdx.x & 31;
  const int hh = row % H;
  const float* hp = h + (size_t)row * C;
  const float* as = a_src + (size_t)hh * C;
  const float* ad = a_dst + (size_t)hh * C;
  float ss = 0.f, sd = 0.f;
  for (int c = lane; c < C; c += 32) {
    const float v = hp[c];
    ss += v * as[c];
    sd += v * ad[c];
  }
#pragma unroll
  for (int off = 16; off > 0; off >>= 1) {
    ss += __shfl_xor(ss, off, 32);
    sd += __shfl_xor(sd, off, 32);
  }
  if (lane == 0) { out_src[row] = ss; out_dst[row] = sd; }
}

__global__ void fill_kernel(float* __restrict__ p, float v, int n) {
  const int i = blockIdx.x * blockDim.x + threadIdx.x;
  if (i < n) p[i] = v;
}

__device__ __forceinline__ void edge_ends(const int* __restrict__ ei, int e,
                                          int& s, int& d) {
  if (e < NEDGES) { s = ei[e]; d = ei[NEDGES + e]; }
  else            { s = d = e - NEDGES; }         // implicit self-loops
}

__device__ __forceinline__ void atomicMaxF(float* addr, float val) {
  int* ai = (int*)addr;
  int cur = __float_as_int(*addr);
  while (__int_as_float(cur) < val) {
    const int assumed = cur;
    cur = atomicCAS(ai, assumed, __float_as_int(val));
    if (cur == assumed) break;
  }
}

// e = leaky_relu(alpha_src[src] + alpha_dst[dst]); segment max over dst
__global__ void edge_logit_max(const int* __restrict__ ei,
    const float* __restrict__ as, const float* __restrict__ ad,
    float* __restrict__ elog, float* __restrict__ mx, int H)
{
  const int idx = blockIdx.x * blockDim.x + threadIdx.x;
  if (idx >= ETOT * H) return;
  const int e = idx / H, hh = idx - e * H;
  int s, d; edge_ends(ei, e, s, d);
  float v = as[s * H + hh] + ad[d * H + hh];
  v = v > 0.f ? v : NEG_SLOPE * v;
  elog[idx] = v;
  atomicMaxF(mx + d * H + hh, v);
}

// e = exp(e - max[dst]); segment sum over dst
__global__ void edge_exp_sum(const int* __restrict__ ei,
    float* __restrict__ elog, const float* __restrict__ mx,
    float* __restrict__ den, int H)
{
  const int idx = blockIdx.x * blockDim.x + threadIdx.x;
  if (idx >= ETOT * H) return;
  const int e = idx / H, hh = idx - e * H;
  int s, d; edge_ends(ei, e, s, d); (void)s;
  const float t = __expf(elog[idx] - mx[d * H + hh]);
  elog[idx] = t;
  atomicAdd(den + d * H + hh, t);
}

// out[dst,h,:] += (e/denom[dst,h]) * hfeat[src,h,:] ; one wave per (edge,head)
// C is a multiple of 128 here (C=512): float4 gathers, 32 lanesx4ch per iter.
__global__ __launch_bounds__(256) void edge_scatter(
    const int* __restrict__ ei, const float* __restrict__ elog,
    const float* __restrict__ den, const float* __restrict__ hfeat,
    float* __restrict__ out, int H, int C)
{
  const int idx = blockIdx.x * 8 + (threadIdx.x >> 5);
  if (idx >= ETOT * H) return;
  const int lane = threadIdx.x & 31;
  const int e = idx / H, hh = idx - e * H;
  int s, d; edge_ends(ei, e, s, d);
  const float alpha = elog[idx] / den[d * H + hh];
  const float* hp = hfeat + ((size_t)s * H + hh) * C;
  float* op = out + ((size_t)d * H + hh) * C;
  for (int c = lane * 4; c < C; c += 128) {
    const float4 hv = *(const float4*)(hp + c);
    atomicAdd(op + c + 0, alpha * hv.x);
    atomicAdd(op + c + 1, alpha * hv.y);
    atomicAdd(op + c + 2, alpha * hv.z);
    atomicAdd(op + c + 3, alpha * hv.w);
  }
}

__global__ void bias_relu(float* __restrict__ x, const float* __restrict__ b,
                          int n, int F) {
  const int i = blockIdx.x * blockDim.x + threadIdx.x;
  if (i < n) {
    const float v = x[i] + b[i % F];
    x[i] = v > 0.f ? v : 0.f;
  }
}

// out[512,10] = g[512,128] @ w[128,10] + b  (tiny, N=10 not WMMA-shaped)
__global__ void fc3_kernel(const float* __restrict__ g,
    const float* __restrict__ w, const float* __restrict__ b,
    float* __restrict__ out)
{
  const int i = blockIdx.x * blockDim.x + threadIdx.x;
  if (i >= 512 * 10) return;
  const int r = i / 10, c = i - r * 10;
  float sum = b[c];
  for (int k = 0; k < 128; ++k) sum += g[r * 128 + k] * w[k * 10 + c];
  out[i] = sum;
}

extern "C" void kernel_launch(void* const* d_in, const int* in_sizes, int n_in,
                              void* d_out, int out_size, void* d_ws, size_t ws_size,
                              hipStream_t stream) {
  (void)in_sizes; (void)n_in; (void)out_size; (void)ws_size;
  const float* x    = (const float*)d_in[0];
  const int*   ei   = (const int*)d_in[1];     // [2, NEDGES]; edge_attr (d_in[2]) unused
  const float* W1   = (const float*)d_in[3];
  const float* a_s1 = (const float*)d_in[4];
  const float* a_d1 = (const float*)d_in[5];
  const float* b1   = (const float*)d_in[6];
  const float* W2   = (const float*)d_in[7];
  const float* a_s2 = (const float*)d_in[8];
  const float* a_d2 = (const float*)d_in[9];
  const float* b2   = (const float*)d_in[10];
  const float* fc1w = (const float*)d_in[11];
  const float* fc1b = (const float*)d_in[12];
  const float* fc2w = (const float*)d_in[13];
  const float* fc2b = (const float*)d_in[14];
  const float* fc3w = (const float*)d_in[15];
  const float* fc3b = (const float*)d_in[16];
  float* out = (float*)d_out;
  float* ws  = (float*)d_ws;

  // --- workspace layout (floats); h1 region is reused for layer2/MLP bufs ---
  float* h1   = ws;                       // 16777216  (4096x4096)
  float* act1 = ws + 16777216;            // 16777216  (4096x4096)
  float* h2   = h1;                       // 2097152   (4096x512)  overlays dead h1
  float* out2 = h1 + 2097152;             // 2097152   (4096x512)
  float* g1   = h1 + 4194304;             // 262144    (512x512)
  float* g2   = g1 + 262144;              // 65536     (512x128)
  float* sm   = ws + 33554432;            // small buffers
  float* as1  = sm;                       // 32768 (4096x8)
  float* ad1  = as1 + 32768;              // 32768
  float* m1   = ad1 + 32768;              // 32768
  float* d1   = m1 + 32768;               // 32768
  float* ea1  = d1 + 32768;               // 294912 (ETOT x 8)
  float* as2  = ea1 + 294912;             // 4096
  float* ad2  = as2 + 4096;               // 4096
  float* m2   = ad2 + 4096;               // 4096
  float* d2   = m2 + 4096;                // 4096
  float* ea2  = d2 + 4096;                // 36864 (ETOT x 1)

  // ======================= GAT layer 1 (H=8, C=512) =======================
  // h1 = x @ W1   [4096,128]x[128,4096]
  gemm_wmma_f32<<<(4096/256)*(4096/64), 256, 0, stream>>>(x, W1, nullptr, h1,
                                                          4096, 4096, 128, 0);
  alpha_kernel<<<32768/8, 256, 0, stream>>>(h1, a_s1, a_d1, as1, ad1, 32768, 8, 512);
  fill_kernel<<<(32768+255)/256, 256, 0, stream>>>(m1, -1e30f, 32768);
  fill_kernel<<<(32768+255)/256, 256, 0, stream>>>(d1, 0.f, 32768);
  fill_kernel<<<(16777216+255)/256, 256, 0, stream>>>(act1, 0.f, 16777216);
  edge_logit_max<<<(ETOT*8+255)/256, 256, 0, stream>>>(ei, as1, ad1, ea1, m1, 8);
  edge_exp_sum  <<<(ETOT*8+255)/256, 256, 0, stream>>>(ei, ea1, m1, d1, 8);
  edge_scatter  <<<(ETOT*8)/8, 256, 0, stream>>>(ei, ea1, d1, h1, act1, 8, 512);
  bias_relu<<<(16777216+255)/256, 256, 0, stream>>>(act1, b1, 16777216, 4096);

  // ======================= GAT layer 2 (H=1, C=512) =======================
  // h2 = act1 @ W2   [4096,4096]x[4096,512]
  gemm_wmma_f32<<<(4096/256)*(512/64), 256, 0, stream>>>(act1, W2, nullptr, h2,
                                                         4096, 512, 4096, 0);
  alpha_kernel<<<4096/8, 256, 0, stream>>>(h2, a_s2, a_d2, as2, ad2, 4096, 1, 512);
  fill_kernel<<<(4096+255)/256, 256, 0, stream>>>(m2, -1e30f, 4096);
  fill_kernel<<<(4096+255)/256, 256, 0, stream>>>(d2, 0.f, 4096);
  fill_kernel<<<(2097152+255)/256, 256, 0, stream>>>(out2, 0.f, 2097152);
  edge_logit_max<<<(ETOT+255)/256, 256, 0, stream>>>(ei, as2, ad2, ea2, m2, 1);
  edge_exp_sum  <<<(ETOT+255)/256, 256, 0, stream>>>(ei, ea2, m2, d2, 1);
  edge_scatter  <<<ETOT/8, 256, 0, stream>>>(ei, ea2, d2, h2, out2, 1, 512);
  bias_relu<<<(2097152+255)/256, 256, 0, stream>>>(out2, b2, 2097152, 512);

  // ============================== MLP head ================================
  // out2 [4096,512] viewed row-major as g [512,4096]
  gemm_wmma_f32<<<(512/256)*(512/64), 256, 0, stream>>>(out2, fc1w, fc1b, g1,
                                                        512, 512, 4096, 1);
  gemm_wmma_f32<<<(512/256)*(128/64), 256, 0, stream>>>(g1, fc2w, fc2b, g2,
                                                        512, 128, 512, 1);
  fc3_kernel<<<(5120+255)/256, 256, 0, stream>>>(g2, fc3w, fc3b, out);
}